// SparseConvNet_48249662603662
// MI455X (gfx1250) — compile-verified
//
#include <hip/hip_runtime.h>
#include <hip/hip_bf16.h>

typedef __attribute__((ext_vector_type(2))) float v2f;
typedef __attribute__((ext_vector_type(8))) float v8f;

#define BN_EPS 1e-3f
#define NPTS 65536

// ---------------------------------------------------------------------------
// mask level 0: m = (mask_u < 0.15)
// ---------------------------------------------------------------------------
__global__ void __launch_bounds__(256)
mask_init_kernel(const float* __restrict__ mu, float* __restrict__ m, int n) {
  int i = blockIdx.x * blockDim.x + threadIdx.x;
  if (i < n) m[i] = (mu[i] < 0.15f) ? 1.0f : 0.0f;
}

// ---------------------------------------------------------------------------
// mask downsample: new_mask = (conv3d(mask, ones3x3x3, stride=2, pad=1) > 0)
// ---------------------------------------------------------------------------
__global__ void __launch_bounds__(256)
mask_down_kernel(const float* __restrict__ in, float* __restrict__ out,
                 int inS, int outS) {
  int n = blockIdx.x * blockDim.x + threadIdx.x;
  int outV = outS * outS * outS;
  if (n >= outV) return;
  int ox = n % outS;
  int t = n / outS;
  int oy = t % outS;
  int oz = t / outS;
  float s = 0.0f;
  for (int dz = -1; dz <= 1; dz++)
    for (int dy = -1; dy <= 1; dy++)
      for (int dx = -1; dx <= 1; dx++) {
        int iz = oz * 2 + dz, iy = oy * 2 + dy, ix = ox * 2 + dx;
        if ((unsigned)iz < (unsigned)inS && (unsigned)iy < (unsigned)inS &&
            (unsigned)ix < (unsigned)inS)
          s += in[(iz * inS + iy) * inS + ix];
      }
  out[n] = (s > 0.0f) ? 1.0f : 0.0f;
}

// ---------------------------------------------------------------------------
// masked input: buf[c*V+v] = x[c*V+v] * m[v]   (V = 64^3 is a power of two)
// ---------------------------------------------------------------------------
__global__ void __launch_bounds__(256)
masked_input_kernel(const float* __restrict__ x, const float* __restrict__ m,
                    float* __restrict__ o, int V, int n) {
  int i = blockIdx.x * blockDim.x + threadIdx.x;
  if (i < n) o[i] = x[i] * m[i & (V - 1)];
}

// ---------------------------------------------------------------------------
// Fused conv3x3x3 (+optional stride 2) + BN + ReLU + mask, implicit GEMM via
// V_WMMA_F32_16X16X4_F32, tap-major ordering:
//   outer: 27 taps (kd,kh,kw) -> per-lane validity + input offset hoisted
//   inner: K = channels, fully unrolled in steps of 4 (CIN is a template arg)
// Layouts (per ISA 7.12.2):
//   A (16x4 f32): lane&15 = M row; lanes<16 carry K={k,k+1}, lanes>=16 K={k+2,k+3}
//   B (4x16 f32): lane&15 = N col; same K split
//   C/D (16x16 f32): vgpr r -> M = 8*(lane>>4)+r, N = lane&15
// ---------------------------------------------------------------------------
template <int CIN>
__global__ void __launch_bounds__(32)
conv_wmma_kernel(const float* __restrict__ in, float* __restrict__ out,
                 const float* __restrict__ wt,
                 const float* __restrict__ bn_g, const float* __restrict__ bn_b,
                 const float* __restrict__ bn_m, const float* __restrict__ bn_v,
                 const float* __restrict__ mask,
                 int inS, int outS, int stride) {
  constexpr int K = CIN * 27;
  const int inV  = inS * inS * inS;
  const int outV = outS * outS * outS;
  const int lane = threadIdx.x;
  const int hi   = lane >> 4;   // 0 or 1 -> K sub-pair
  const int lo   = lane & 15;

  const int coTile = blockIdx.y;
  const int co     = coTile * 16 + lo;       // A-matrix row (out channel)
  const int n      = blockIdx.x * 16 + lo;   // B-matrix col (voxel)

  // decode output voxel -> input base coordinates
  const int ox = n % outS;
  const int t1 = n / outS;
  const int oy = t1 % outS;
  const int oz = t1 / outS;
  const int bx = ox * stride - 1;
  const int by = oy * stride - 1;
  const int bz = oz * stride - 1;
  const int base = stride * ((oz * inS + oy) * inS + ox);

  const float* __restrict__ wrow = wt + co * K + hi * 2 * 27;  // K pair base
  const int kkV = hi * 2;  // channel sub-offset for B

  v8f acc = {0.f, 0.f, 0.f, 0.f, 0.f, 0.f, 0.f, 0.f};

  for (int kd = 0; kd < 3; kd++) {
    const bool vz = (unsigned)(bz + kd) < (unsigned)inS;
    for (int kh = 0; kh < 3; kh++) {
      const bool vy = (unsigned)(by + kh) < (unsigned)inS;
      for (int kw = 0; kw < 3; kw++) {
        const bool valid =
            vz && vy && ((unsigned)(bx + kw) < (unsigned)inS);
        const int dOff = (kd - 1) * inS * inS + (kh - 1) * inS + (kw - 1);
        // clamp address for invalid lanes (stays in-bounds), zero via multiply
        const int   sbase = valid ? base + dOff : 0;
        const float vf    = valid ? 1.0f : 0.0f;
        const int   tap   = kd * 9 + kh * 3 + kw;

        const float* __restrict__ wp = wrow + tap;          // A loads
        const float* __restrict__ ip = in + sbase + kkV * inV;  // B loads

#pragma unroll
        for (int c4 = 0; c4 < CIN; c4 += 4) {
          v2f av, bv;
          av.x = wp[c4 * 27];            // W[co][c4+kk  ][tap]
          av.y = wp[(c4 + 1) * 27];      // W[co][c4+kk+1][tap]
          bv.x = ip[c4 * inV] * vf;      // In[c4+kk  ][voxel+tap]
          bv.y = ip[(c4 + 1) * inV] * vf;
          acc = __builtin_amdgcn_wmma_f32_16x16x4_f32(
              /*neg_a=*/false, av, /*neg_b=*/false, bv,
              /*c_mod=*/(short)0, acc, /*reuse_a=*/false, /*reuse_b=*/false);
        }
      }
    }
  }

  // fused BN + ReLU + mask epilogue
  const float mval = mask[n];
#pragma unroll
  for (int r = 0; r < 8; r++) {
    const int c = coTile * 16 + hi * 8 + r;
    const float s = bn_g[c] / sqrtf(bn_v[c] + BN_EPS);
    float val = acc[r] * s + (bn_b[c] - bn_m[c] * s);
    val = fmaxf(val, 0.0f);
    out[c * outV + n] = val * mval;
  }
}

// ---------------------------------------------------------------------------
// trilinear grid sample: out[p*192 + base + c], matches reference exactly
// ---------------------------------------------------------------------------
__global__ void __launch_bounds__(256)
grid_sample_kernel(const float* __restrict__ vol, const float* __restrict__ coords,
                   float* __restrict__ out, int C, int S, int outBase) {
  int p = blockIdx.x * blockDim.x + threadIdx.x;
  if (p >= NPTS) return;

  const float gx = coords[p * 3 + 0];
  const float gy = coords[p * 3 + 1];
  const float gz = coords[p * 3 + 2];
  const float ix = (gx + 1.0f) * 0.5f * (float)(S - 1);
  const float iy = (gy + 1.0f) * 0.5f * (float)(S - 1);
  const float iz = (gz + 1.0f) * 0.5f * (float)(S - 1);
  const float fx = floorf(ix), fy = floorf(iy), fz = floorf(iz);
  const int x0 = (int)fx, y0 = (int)fy, z0 = (int)fz;
  const float wx = ix - fx, wy = iy - fy, wz = iz - fz;

  int   idx8[8];
  float w8[8];
  int t = 0;
  for (int dz = 0; dz <= 1; dz++)
    for (int dy = 0; dy <= 1; dy++)
      for (int dx = 0; dx <= 1; dx++) {
        int zi = z0 + dz, yi = y0 + dy, xi = x0 + dx;
        bool valid = ((unsigned)zi < (unsigned)S) &&
                     ((unsigned)yi < (unsigned)S) &&
                     ((unsigned)xi < (unsigned)S);
        int zc = zi < 0 ? 0 : (zi > S - 1 ? S - 1 : zi);
        int yc = yi < 0 ? 0 : (yi > S - 1 ? S - 1 : yi);
        int xc = xi < 0 ? 0 : (xi > S - 1 ? S - 1 : xi);
        idx8[t] = (zc * S + yc) * S + xc;
        float w = (dx ? wx : 1.0f - wx) * (dy ? wy : 1.0f - wy) *
                  (dz ? wz : 1.0f - wz);
        w8[t] = valid ? w : 0.0f;
        t++;
      }

  const int V = S * S * S;
  for (int c = 0; c < C; c++) {
    const float* vc = vol + c * V;
    float acc = 0.0f;
#pragma unroll
    for (int q = 0; q < 8; q++) acc += vc[idx8[q]] * w8[q];
    out[p * 192 + outBase + c] = acc;
  }
}

// ---------------------------------------------------------------------------
// host-side launch
// ---------------------------------------------------------------------------
extern "C" void kernel_launch(void* const* d_in, const int* in_sizes, int n_in,
                              void* d_out, int out_size, void* d_ws, size_t ws_size,
                              hipStream_t stream) {
  (void)out_size; (void)ws_size;

  // --- identify x / mask_u / coords by unique sizes, collect param leaves ---
  const float* x      = nullptr;
  const float* mask_u = nullptr;
  const float* coords = nullptr;
  int pidx[80];
  int np = 0;
  for (int i = 0; i < n_in; i++) {
    const int s = in_sizes[i];
    if (s == 16 * 64 * 64 * 64 && !x)      x      = (const float*)d_in[i];
    else if (s == 64 * 64 * 64 && !mask_u) mask_u = (const float*)d_in[i];
    else if (s == NPTS * 3 && !coords)     coords = (const float*)d_in[i];
    else if (np < 80)                      pidx[np++] = i;
  }
  if (!x || !mask_u || !coords || np < 65) return;  // unexpected layout

  // --- pytree ordering: insertion (conv0,down0,conv1,down1,conv2,down2,conv3)
  //     vs sorted keys (conv0..conv3,down0..down2).  Discriminator: params
  //     leaf #25 is down1.w (55296) only in insertion order. ---
  const bool sortedOrder = (in_sizes[pidx[25]] != 55296);
  const int l2g_srt[13] = {0, 1, 10, 2, 3, 11, 4, 5, 6, 12, 7, 8, 9};

  struct Layer { const float *w, *g, *b, *m, *v; };
  Layer L[13];
  for (int l = 0; l < 13; l++) {
    const int gi = sortedOrder ? l2g_srt[l] : l;
    const int bs = gi * 5;
    L[l].w = (const float*)d_in[pidx[bs + 0]];
    L[l].g = (const float*)d_in[pidx[bs + 1]];
    L[l].b = (const float*)d_in[pidx[bs + 2]];
    L[l].m = (const float*)d_in[pidx[bs + 3]];
    L[l].v = (const float*)d_in[pidx[bs + 4]];
  }

  // --- workspace layout (floats) ---
  float* ws   = (float*)d_ws;
  float* m0   = ws;                 // 262144
  float* m1   = m0 + 262144;        // 32768
  float* m2   = m1 + 32768;         // 4096
  float* m3   = m2 + 4096;          // 512
  float* bufA = m3 + 1024;          // 32*64^3 = 8388608
  float* bufB = bufA + 8388608;     // 8388608

  float* out = (float*)d_out;

  // --- masks at all resolutions ---
  mask_init_kernel<<<(262144 + 255) / 256, 256, 0, stream>>>(mask_u, m0, 262144);
  mask_down_kernel<<<(32768 + 255) / 256, 256, 0, stream>>>(m0, m1, 64, 32);
  mask_down_kernel<<<(4096 + 255) / 256, 256, 0, stream>>>(m1, m2, 32, 16);
  mask_down_kernel<<<(512 + 255) / 256, 256, 0, stream>>>(m2, m3, 16, 8);

  // --- masked input into bufA ---
  masked_input_kernel<<<(16 * 262144 + 255) / 256, 256, 0, stream>>>(
      x, m0, bufA, 262144, 16 * 262144);

  // --- conv pipeline ---
  const int   Cin[13]  = {16, 32, 32, 32, 32, 32, 64, 64, 64, 64, 96, 96, 96};
  const int   Cout[13] = {32, 32, 32, 32, 32, 64, 64, 64, 64, 96, 96, 96, 96};
  const int   inS[13]  = {64, 64, 64, 32, 32, 32, 16, 16, 16, 16, 8, 8, 8};
  const int   outS[13] = {64, 64, 32, 32, 32, 16, 16, 16, 16, 8, 8, 8, 8};
  const int   strd[13] = {1, 1, 2, 1, 1, 2, 1, 1, 1, 2, 1, 1, 1};
  float* const msk[13] = {m0, m0, m1, m1, m1, m2, m2, m2, m2, m3, m3, m3, m3};

  float* src = bufA;
  float* dst = bufB;
  for (int l = 0; l < 13; l++) {
    const int oV = outS[l] * outS[l] * outS[l];
    dim3 grid(oV / 16, Cout[l] / 16);
    switch (Cin[l]) {
      case 16:
        conv_wmma_kernel<16><<<grid, 32, 0, stream>>>(
            src, dst, L[l].w, L[l].g, L[l].b, L[l].m, L[l].v, msk[l],
            inS[l], outS[l], strd[l]);
        break;
      case 32:
        conv_wmma_kernel<32><<<grid, 32, 0, stream>>>(
            src, dst, L[l].w, L[l].g, L[l].b, L[l].m, L[l].v, msk[l],
            inS[l], outS[l], strd[l]);
        break;
      case 64:
        conv_wmma_kernel<64><<<grid, 32, 0, stream>>>(
            src, dst, L[l].w, L[l].g, L[l].b, L[l].m, L[l].v, msk[l],
            inS[l], outS[l], strd[l]);
        break;
      default:
        conv_wmma_kernel<96><<<grid, 32, 0, stream>>>(
            src, dst, L[l].w, L[l].g, L[l].b, L[l].m, L[l].v, msk[l],
            inS[l], outS[l], strd[l]);
        break;
    }

    if (l == 4)   // f1: 32 channels @ 32^3
      grid_sample_kernel<<<NPTS / 256, 256, 0, stream>>>(dst, coords, out, 32, 32, 0);
    if (l == 8)   // f2: 64 channels @ 16^3
      grid_sample_kernel<<<NPTS / 256, 256, 0, stream>>>(dst, coords, out, 64, 16, 32);
    if (l == 12)  // f3: 96 channels @ 8^3
      grid_sample_kernel<<<NPTS / 256, 256, 0, stream>>>(dst, coords, out, 96, 8, 96);

    float* tmp = src; src = dst; dst = tmp;
  }
}